// NST_82162724372481
// MI455X (gfx1250) — compile-verified
//
#include <hip/hip_runtime.h>
#include <cstdint>

// Soft decision forest, DEPTH=10, 32 trees, batch 4096.
// Memory-bound: 536 MB in / 512 KB out -> ~23 us floor at 23.3 TB/s.
// Strategy: one wave32 per (batch,tree) pair; async global->LDS staging
// (global_load_async_to_lds_b128, ASYNCcnt) double-buffered per wave;
// per-lane subtree evaluation; shuffle reduction.

#define DEPTH        10
#define NUM_TREES    32
#define BATCH        4096
#define NUM_FORKS    1023                     // 2^10 - 1
#define PAIRS        (BATCH * NUM_TREES)      // 131072
#define PAIR_BYTES   (NUM_FORKS * 4)          // 4092

#define BLOCK_THREADS   128
#define WAVES_PER_BLOCK (BLOCK_THREADS / 32)  // 4
#define GRID_BLOCKS     2048
#define TOTAL_WAVES     (GRID_BLOCKS * WAVES_PER_BLOCK)  // 8192
#define PAIRS_PER_WAVE  (PAIRS / TOTAL_WAVES)            // 16 exactly

__device__ __forceinline__ float fast_sigmoid(float z) {
    // sigmoid(z) = 1 / (1 + 2^(-z*log2(e)))  -> v_exp_f32 + v_rcp_f32
    return __builtin_amdgcn_rcpf(1.0f + __builtin_amdgcn_exp2f(-1.4426950408889634f * z));
}

// Issue async copy of one pair's 1023 floats into an LDS buffer.
// Returns the number of ASYNC ops issued (8 normally, 7 for the global-last
// pair, which finishes its tail with guarded scalar loads to avoid OOB).
__device__ __forceinline__ int issue_pair_copy(const float* __restrict__ x,
                                               float* __restrict__ ldsBuf,
                                               int pair, int lane) {
    unsigned long long ga = (unsigned long long)(uintptr_t)x
                          + (unsigned long long)(unsigned)pair * (unsigned long long)PAIR_BYTES
                          + (unsigned)(lane * 16);
    // Low 32 bits of the generic (flat) LDS address == workgroup-relative LDS
    // byte offset (ISA 10.2: LDS aperture).
    unsigned la = (unsigned)(uintptr_t)ldsBuf + (unsigned)(lane * 16);

    if (pair != PAIRS - 1) {
        // 8 x b128: lanes cover 512 B per op, instruction offset advances BOTH
        // the global and LDS address (ISA 8.async: INST_OFFSET added to both).
        asm volatile(
            "global_load_async_to_lds_b128 %0, %1, off\n\t"
            "global_load_async_to_lds_b128 %0, %1, off offset:2048\n\t"
            "global_load_async_to_lds_b128 %0, %1, off offset:4096\n\t"
            "global_load_async_to_lds_b128 %0, %1, off offset:6144\n\t"
            "global_load_async_to_lds_b128 %0, %1, off offset:8192\n\t"
            "global_load_async_to_lds_b128 %0, %1, off offset:10240\n\t"
            "global_load_async_to_lds_b128 %0, %1, off offset:12288\n\t"
            "global_load_async_to_lds_b128 %0, %1, off offset:14336"
            :: "v"(la), "v"(ga) : "memory");
        return 8;
    } else {
        // Last pair of the whole problem: 7 async ops cover elements [512j,512j+128) for j<7;
        // guarded scalar tail covers [896,1023) without reading past x.
        asm volatile(
            "global_load_async_to_lds_b128 %0, %1, off\n\t"
            "global_load_async_to_lds_b128 %0, %1, off offset:2048\n\t"
            "global_load_async_to_lds_b128 %0, %1, off offset:4096\n\t"
            "global_load_async_to_lds_b128 %0, %1, off offset:6144\n\t"
            "global_load_async_to_lds_b128 %0, %1, off offset:8192\n\t"
            "global_load_async_to_lds_b128 %0, %1, off offset:10240\n\t"
            "global_load_async_to_lds_b128 %0, %1, off offset:12288"
            :: "v"(la), "v"(ga) : "memory");
        const float* src = x + (size_t)pair * NUM_FORKS;
        for (int e = 896 + lane; e < NUM_FORKS; e += 32) ldsBuf[e] = src[e];
        return 7;
    }
}

__device__ __forceinline__ float eval_pair(const float* __restrict__ b, int lane) {
    // Path product for this lane's level-5 subtree root (levels 0..4).
    float p = 1.0f;
#pragma unroll
    for (int i = 0; i < 5; ++i) {
        float z = b[(1 << i) - 1 + (lane >> (5 - i))];
        float s = fast_sigmoid(z * (1.0f / (float)(1 << (9 - i))));
        p *= ((lane >> (4 - i)) & 1) ? s : (1.0f - s);
    }
    // Expand through levels 5..8: 1 -> 16 reach-probabilities in registers.
    float pr[16];
    pr[0] = p;
#pragma unroll
    for (int d = 0; d < 4; ++d) {
        const int lev  = 5 + d;
        const int base = (1 << lev) - 1 + (lane << d);
        const float sc = 1.0f / (float)(1 << (9 - lev));
#pragma unroll
        for (int m = (1 << d) - 1; m >= 0; --m) {
            float s  = fast_sigmoid(b[base + m] * sc);
            float pm = pr[m];
            pr[2 * m + 1] = pm * s;
            pr[2 * m]     = pm - pm * s;       // pm*(1-s), FMA-friendly
        }
    }
    // Level 9 (scale = 1): weighted sum of right-branch sigmoids.
    float acc = 0.0f;
    const int base9 = 511 + (lane << 4);
#pragma unroll
    for (int m = 0; m < 16; ++m)
        acc += pr[m] * fast_sigmoid(b[base9 + m]);
    return acc;
}

__global__ __launch_bounds__(BLOCK_THREADS)
void soft_forest_kernel(const float* __restrict__ x, float* __restrict__ out) {
    __shared__ float smem[WAVES_PER_BLOCK * 2 * 1024];   // 32 KB: 2 bufs / wave

    const int lane  = threadIdx.x & 31;
    const int winb  = threadIdx.x >> 5;
    const int gwave = blockIdx.x * WAVES_PER_BLOCK + winb;

    float* buf0 = smem + winb * 2048;
    float* buf1 = buf0 + 1024;

    int pair = gwave;
    int sel  = 0;
    issue_pair_copy(x, buf0, pair, lane);          // prefetch first pair

#pragma unroll 1
    for (int k = 0; k < PAIRS_PER_WAVE; ++k) {
        const int nxt = pair + TOTAL_WAVES;
        int nIssued = 0;
        if (k + 1 < PAIRS_PER_WAVE)
            nIssued = issue_pair_copy(x, (sel ? buf0 : buf1), nxt, lane);

        // Wait until only the just-issued (next-buffer) ops remain in flight:
        // async ops complete in order, so current buffer is then resident.
        if (nIssued == 8)      asm volatile("s_wait_asynccnt 0x8" ::: "memory");
        else if (nIssued == 7) asm volatile("s_wait_asynccnt 0x7" ::: "memory");
        else                   asm volatile("s_wait_asynccnt 0x0" ::: "memory");

        float acc = eval_pair(sel ? buf1 : buf0, lane);

        // Butterfly reduction across the wave (wave32).
        acc += __shfl_xor(acc, 16);
        acc += __shfl_xor(acc, 8);
        acc += __shfl_xor(acc, 4);
        acc += __shfl_xor(acc, 2);
        acc += __shfl_xor(acc, 1);
        if (lane == 0) out[pair] = acc;

        pair = nxt;
        sel ^= 1;
    }
}

extern "C" void kernel_launch(void* const* d_in, const int* in_sizes, int n_in,
                              void* d_out, int out_size, void* d_ws, size_t ws_size,
                              hipStream_t stream) {
    const float* x = (const float*)d_in[0];
    float* out = (float*)d_out;
    soft_forest_kernel<<<GRID_BLOCKS, BLOCK_THREADS, 0, stream>>>(x, out);
}